// Frames_18837726560432
// MI455X (gfx1250) — compile-verified
//
#include <hip/hip_runtime.h>

// Problem dimensions (fixed by the reference).
#define B_ROWS   4096
#define W_ENC    8192
#define MAX_LEN  4096
#define TPB      256                            // 8 wave32 waves per block
#define GROUPS   8                              // float4 groups per thread
#define GROUP_FLOATS   (TPB * 4)                // 1024 floats per group window
#define BLOCK_FLOATS   (GROUP_FLOATS * GROUPS)  // 8192 = one full row per block

typedef float v4f  __attribute__((ext_vector_type(4)));              // 16B-aligned
typedef float v4fu __attribute__((ext_vector_type(4), aligned(4)));  // element-aligned

// out_y[b, j] = (j < 8192-L) ? prev[b, L+j] : ragged[b, j-(8192-L)],  L = lens[b]
// out_lens[b] = (float)L
//
// One block per row. Per group the window [gbase, gbase+1024) is compared
// against the (block-uniform, SGPR-resident) split point, so path selection is
// a scalar s_cmp/s_cbranch -- no exec-mask divergence machinery. Fast paths
// are one NT global_load_b128
// + one aligned NT global_store_b128, with group offsets folded
// into the 24-bit instruction immediate. At most one window per row straddles
// the split and takes the per-element select path.
__global__ __launch_bounds__(TPB) void ragged_shift_gather(
    const float* __restrict__ prev,       // [B_ROWS, W_ENC]
    const float* __restrict__ rag,        // [B_ROWS, MAX_LEN]
    const int*   __restrict__ lens,       // [B_ROWS]
    float*       __restrict__ out_y,      // [B_ROWS, W_ENC]
    float*       __restrict__ out_lens)   // [B_ROWS]
{
    const int b     = (int)blockIdx.x;            // row
    const int L     = lens[b];                    // shift, 0..4095 (scalar load)
    const int split = W_ENC - L;                  // first index sourced from ragged

    const float* __restrict__ prow = prev + (size_t)b * W_ENC + L;  // prow[j], valid j < split
    const float* __restrict__ rrow = rag  + (size_t)b * MAX_LEN;    // rrow[j-split], j >= split
    float*       __restrict__ orow = out_y + (size_t)b * W_ENC;

    const int t4 = (int)threadIdx.x * 4;          // thread's float offset in each window

#pragma unroll
    for (int g = 0; g < GROUPS; ++g) {
        const int gbase = g * GROUP_FLOATS;       // block-uniform window base
        const int j0    = gbase + t4;
        v4f v;
        if (split >= gbase + GROUP_FLOATS) {
            // Whole window from prev (scalar branch; uniform across block).
            v = __builtin_nontemporal_load((const v4fu*)(prow + j0));
        } else if (split <= gbase) {
            // Whole window from ragged.
            v = __builtin_nontemporal_load((const v4fu*)(rrow + (j0 - split)));
        } else {
            // split lies inside this window: per-element branchless select.
#pragma unroll
            for (int e = 0; e < 4; ++e) {
                const int j = j0 + e;
                const float* src = (j < split) ? (prow + j) : (rrow + (j - split));
                v[e] = __builtin_nontemporal_load(src);
            }
        }
        // Output offset is a multiple of 16B within a 32KiB row: aligned b128 NT store.
        __builtin_nontemporal_store(v, (v4f*)(orow + j0));
    }

    // Second tuple output: lens (one value per row).
    if (threadIdx.x == 0) {
        out_lens[b] = (float)L;
    }
}

extern "C" void kernel_launch(void* const* d_in, const int* in_sizes, int n_in,
                              void* d_out, int out_size, void* d_ws, size_t ws_size,
                              hipStream_t stream) {
    const float* prev = (const float*)d_in[0];   // (4096, 8192) f32
    const float* rag  = (const float*)d_in[1];   // (4096, 4096) f32
    const int*   lens = (const int*)  d_in[2];   // (4096,) i32

    float* out_y    = (float*)d_out;                             // 4096*8192 floats
    float* out_lens = (float*)d_out + (size_t)B_ROWS * W_ENC;    // then 4096 lens

    const dim3 grid(B_ROWS);                     // one block per row
    const dim3 block(TPB);
    ragged_shift_gather<<<grid, block, 0, stream>>>(prev, rag, lens, out_y, out_lens);
}